// RawISPProcessing_75737453298025
// MI455X (gfx1250) — compile-verified
//
#include <hip/hip_runtime.h>
#include <cstdint>

// ---------------- problem constants ----------------
#define BATCH   8
#define H_IN    512
#define W_IN    512
#define H_OUT   1024
#define W_OUT   1024

// ---------------- tiling ----------------
#define TW_OUT  256          // output tile width
#define TH_OUT  16           // output tile height
#define TW_IN   128          // input tile core width
#define TH_IN   8            // input tile core height
#define LDS_ROWS 10          // TH_IN + 2 halo
#define LDS_USED 130         // TW_IN + 2 halo
#define LDS_COLS 132         // padded stride
#define ELEMS_PER_CH (LDS_ROWS * LDS_USED)   // 1300
#define TOT_ELEMS    (4 * ELEMS_PER_CH)      // 5200

typedef __attribute__((ext_vector_type(2))) float v2f;
typedef __attribute__((ext_vector_type(8))) float v8f;

// Per-batch fused color matrix on the WMMA pipe (exact, f32):
//   M2[b][k][e] = 2 * sum_c awb_b[2-e][c] * ccm_b[c][2-k]
// i.e. 8 batched 3x3 GEMMs = the only real matrix product in the pipeline.
// We use V_WMMA_F32_16X16X4_F32 twice per batch, placing contraction terms
// only in K=0 (A/B VGPR0, lanes 0-15) and K=3 (A/B VGPR1, lanes 16-31),
// which are layout-invariant slots; c=2 is accumulated through the C operand.
__global__ __launch_bounds__(32)
void isp_matrix_precompute_wmma(const float* __restrict__ awb,
                                const float* __restrict__ ccm,
                                float* __restrict__ M) {
    const int lane = threadIdx.x;        // exactly one wave32, EXEC all ones
    const int half = lane >> 4;          // 0: lanes 0-15, 1: lanes 16-31
    const int lm   = lane & 15;          // row (A/M) or col (B/N) within half

    #pragma unroll 1
    for (int b = 0; b < BATCH; ++b) {
        const float* A9 = awb + b * 9;   // awb[row*3+col]
        const float* C9 = ccm + b * 9;

        // Left operand  X[k][c] = 2*ccm[c][2-k]   (k = output row)
        // Right operand Y[c][e] = awb[2-e][c]     (e = output col)
        v2f a1 = {0.0f, 0.0f}, b1 = {0.0f, 0.0f};
        v2f a2 = {0.0f, 0.0f}, b2 = {0.0f, 0.0f};
        if (lm < 3) {
            if (half == 0) {
                // K=0 slot: A VGPR0 lanes 0-15 / B VGPR0 lanes 0-15
                a1.x = 2.0f * C9[0 * 3 + (2 - lm)];   // X[lm][0]
                b1.x = A9[(2 - lm) * 3 + 0];          // Y[0][lm]
                a2.x = 2.0f * C9[2 * 3 + (2 - lm)];   // X[lm][2]
                b2.x = A9[(2 - lm) * 3 + 2];          // Y[2][lm]
            } else {
                // K=3 slot: A VGPR1 lanes 16-31 / B VGPR1 lanes 16-31
                a1.y = 2.0f * C9[1 * 3 + (2 - lm)];   // X[lm][1]
                b1.y = A9[(2 - lm) * 3 + 1];          // Y[1][lm]
            }
        }
        v8f c0 = {};
        // D1[m][n] = X[m][0]*Y[0][n] + X[m][1]*Y[1][n]
        v8f d1 = __builtin_amdgcn_wmma_f32_16x16x4_f32(
                     false, a1, false, b1, (short)0, c0, false, false);
        // D2[m][n] = D1[m][n] + X[m][2]*Y[2][n]  == M2[b][k=m][e=n]
        v8f d2 = __builtin_amdgcn_wmma_f32_16x16x4_f32(
                     false, a2, false, b2, (short)0, d1, false, false);

        // C/D layout: VGPR r holds row M=r on lanes 0-15, column N=lane.
        if (lane < 3) {
            M[b * 9 + 0 * 3 + lane] = d2[0];
            M[b * 9 + 1 * 3 + lane] = d2[1];
            M[b * 9 + 2 * 3 + lane] = d2[2];
        }
    }
}

__global__ __launch_bounds__(256)
void isp_main(const float* __restrict__ pred,
              const float* __restrict__ gt,
              const float* __restrict__ M,
              float* __restrict__ out) {
    // 4 channels x (8+2 halo) rows x 130 cols (stride 132) = 21120 B LDS
    __shared__ float lds[4 * LDS_ROWS * LDS_COLS];

    const int  tz = blockIdx.z;           // b*2 + src_select
    const int  b  = tz >> 1;
    const int  s  = tz & 1;
    const float* src = (s ? gt : pred) + (size_t)b * 4 * H_IN * W_IN;
    float* dst = out + (size_t)s * ((size_t)BATCH * 3 * H_OUT * W_OUT)
                     + (size_t)b * 3 * H_OUT * W_OUT;

    const int ch0 = blockIdx.y * TH_IN;   // first core input row
    const int cw0 = blockIdx.x * TW_IN;   // first core input col
    const int tid = threadIdx.x;

    // ---- stage input tile (+clamped halo) into LDS via CDNA5 async DMA ----
    #pragma unroll 1
    for (int base = 0; base < TOT_ELEMS; base += 256) {
        int e = base + tid;
        e = (e < TOT_ELEMS) ? e : (TOT_ELEMS - 1);   // tail lanes duplicate (benign)
        int ch  = e / ELEMS_PER_CH;
        int rem = e - ch * ELEMS_PER_CH;
        int r   = rem / LDS_USED;
        int c   = rem - r * LDS_USED;
        int gh = ch0 - 1 + r; gh = gh < 0 ? 0 : (gh > H_IN - 1 ? H_IN - 1 : gh);
        int gw = cw0 - 1 + c; gw = gw < 0 ? 0 : (gw > W_IN - 1 ? W_IN - 1 : gw);
        const float* gp = src + ((size_t)ch * H_IN + gh) * W_IN + gw;
        // low 32 bits of a generic LDS pointer == byte offset within the
        // wave's LDS allocation, which is what the async VDST operand expects.
        unsigned lo = (unsigned)(uintptr_t)(&lds[(ch * LDS_ROWS + r) * LDS_COLS + c]);
        asm volatile("global_load_async_to_lds_b32 %0, %1, off"
                     :: "v"(lo), "v"((unsigned long long)(uintptr_t)gp)
                     : "memory");
    }
    asm volatile("s_wait_asynccnt 0" ::: "memory");
    __syncthreads();

    // ---- per-batch fused color matrix (uniform loads) ----
    const float* Mb = M + b * 9;
    const float m00 = Mb[0], m01 = Mb[1], m02 = Mb[2];
    const float m10 = Mb[3], m11 = Mb[4], m12 = Mb[5];
    const float m20 = Mb[6], m21 = Mb[7], m22 = Mb[8];

    // ---- each thread: 4x4 output patch (2x2 input footprint + halo) ----
    const int tx  = tid & 63;     // 64 threads across width
    const int ty  = tid >> 6;     // 4 threads across height
    const int lr0 = ty * 2;       // first LDS tap row (input row hh0-1)
    const int lc0 = tx * 2;       // first LDS tap col (input col ww0-1)

    float T[4][4][4];             // [channel][tap_row][tap_col]
    #pragma unroll
    for (int ch = 0; ch < 4; ++ch)
        #pragma unroll
        for (int r = 0; r < 4; ++r)
            #pragma unroll
            for (int c = 0; c < 4; ++c)
                T[ch][r][c] = lds[(ch * LDS_ROWS + lr0 + r) * LDS_COLS + lc0 + c];

    // half-pixel bilinear 2x up: output subrow ii uses tap pair (ta, ta+1)
    // with weights (wa, 1-wa); same table for columns.
    const int   ta[4] = {0, 1, 1, 2};
    const float wa[4] = {0.25f, 0.75f, 0.25f, 0.75f};
    const float invg  = 1.0f / 2.2f;

    const int i0 = blockIdx.y * TH_OUT + ty * 4;
    const int j0 = blockIdx.x * TW_OUT + tx * 4;

    #pragma unroll
    for (int ii = 0; ii < 4; ++ii) {
        const int   rA  = ta[ii];
        const float wrA = wa[ii], wrB = 1.0f - wa[ii];
        float o[3][4];
        #pragma unroll
        for (int jj = 0; jj < 4; ++jj) {
            const int   cA  = ta[jj];
            const float wcA = wa[jj], wcB = 1.0f - wa[jj];
            float bil[4];
            #pragma unroll
            for (int ch = 0; ch < 4; ++ch) {
                float ra = T[ch][rA    ][cA] * wcA + T[ch][rA    ][cA + 1] * wcB;
                float rb = T[ch][rA + 1][cA] * wcA + T[ch][rA + 1][cA + 1] * wcB;
                bil[ch] = ra * wrA + rb * wrB;
            }
            // green parity select (flips in reference are identities for
            // symmetric half-pixel bilinear 2x upsampling)
            float G = ((ii ^ jj) & 1) ? ((ii & 1) ? bil[2] : bil[1])
                                      : 0.5f * (bil[1] + bil[2]);
            float v0 = bil[0], v1 = G, v2 = bil[3];
            float t0 = fmaxf(fmaf(m00, v0, fmaf(m01, v1, m02 * v2)), 1e-8f);
            float t1 = fmaxf(fmaf(m10, v0, fmaf(m11, v1, m12 * v2)), 1e-8f);
            float t2 = fmaxf(fmaf(m20, v0, fmaf(m21, v1, m22 * v2)), 1e-8f);
            // pow(t, 1/2.2) = 2^((1/2.2)*log2(t)); t >= 1e-8 > 0, maps to
            // hardware v_log_f32 (log2) + v_exp_f32 (exp2).
            o[0][jj] = __builtin_amdgcn_exp2f(invg * __builtin_amdgcn_logf(t0));
            o[1][jj] = __builtin_amdgcn_exp2f(invg * __builtin_amdgcn_logf(t1));
            o[2][jj] = __builtin_amdgcn_exp2f(invg * __builtin_amdgcn_logf(t2));
        }
        const int i = i0 + ii;
        #pragma unroll
        for (int k = 0; k < 3; ++k) {
            float4 v = make_float4(o[k][0], o[k][1], o[k][2], o[k][3]);
            *reinterpret_cast<float4*>(dst + ((size_t)k * H_OUT + i) * W_OUT + j0) = v;
        }
    }
}

extern "C" void kernel_launch(void* const* d_in, const int* in_sizes, int n_in,
                              void* d_out, int out_size, void* d_ws, size_t ws_size,
                              hipStream_t stream) {
    const float* pred = (const float*)d_in[0];   // (8,4,512,512)
    const float* gt   = (const float*)d_in[1];   // (8,4,512,512)
    const float* awb  = (const float*)d_in[2];   // (8,3,3)
    const float* ccm  = (const float*)d_in[3];   // (8,3,3)
    // d_in[4] = rgb_gain: unused by the reference computation
    float* M   = (float*)d_ws;                   // 72 floats of scratch
    float* out = (float*)d_out;                  // x (25,165,824 f32) then y

    hipLaunchKernelGGL(isp_matrix_precompute_wmma, dim3(1), dim3(32), 0, stream,
                       awb, ccm, M);

    dim3 grid(W_OUT / TW_OUT, H_OUT / TH_OUT, BATCH * 2);   // (4, 64, 16)
    hipLaunchKernelGGL(isp_main, grid, dim3(256), 0, stream,
                       pred, gt, M, out);
}